// OSG_C_29987461660961
// MI455X (gfx1250) — compile-verified
//
#include <hip/hip_runtime.h>
#include <math.h>

// ---------------------------------------------------------------------------
// Problem constants (from reference): B=2, N=2048, F=1024, H=512, E=256, K=30
// ---------------------------------------------------------------------------
#define BB   2
#define NSEQ 2048
#define FF   1024
#define HH   512
#define EE   256
#define KSEG 30
#define MROWS (BB * NSEQ)   // 4096
#define BIGF 1e30f

typedef float v2f __attribute__((ext_vector_type(2)));
typedef float v8f __attribute__((ext_vector_type(8)));

// ---------------------------------------------------------------------------
// f32 WMMA tile: one wave computes a 16x16 tile of  A[M,K] * B[Ncol,K]^T
// using V_WMMA_F32_16X16X4_F32.
// ISA f32 layouts (CDNA5 ISA):
//   A 16x4 : lane&15 = m, lane>>4 selects K-pair {0,1}/{2,3}; v0=K, v1=K+1
//   B 4x16 : lane&15 = n, lane>>4 selects K-pair;             v0=K, v1=K+1
//   C 16x16: lane&15 = n, row m = 8*(lane>>4) + r  for VGPR r
// Both operands are rows along K -> aligned float2 loads.
// ---------------------------------------------------------------------------
__device__ __forceinline__ v8f wmma_dotT_f32(const float* __restrict__ A,
                                             const float* __restrict__ B,
                                             int K, int row0, int col0) {
  const int lane = threadIdx.x & 31;
  const int half = lane >> 4;   // selects K pair {0,1} vs {2,3}
  const int l    = lane & 15;   // m for A, n for B
  const float* ap = A + (size_t)(row0 + l) * K + 2 * half;
  const float* bp = B + (size_t)(col0 + l) * K + 2 * half;
  v8f acc = {};
  for (int k = 0; k < K; k += 4) {
    v2f a = *(const v2f*)(ap + k);
    v2f b = *(const v2f*)(bp + k);
    acc = __builtin_amdgcn_wmma_f32_16x16x4_f32(false, a, false, b,
                                                (short)0, acc, false, false);
  }
  return acc;
}

// C[M,Ncol] = act(A[M,K] @ Bw[Ncol,K]^T + bias),  act = relu if relu!=0
__global__ void k_gemm(const float* __restrict__ A, const float* __restrict__ Bw,
                       const float* __restrict__ bias, float* __restrict__ C,
                       int M, int Ncol, int K, int relu) {
  const int wave = threadIdx.x >> 5;
  const int tN = blockIdx.x * 8 + wave;   // 16-wide column tile
  const int tM = blockIdx.y;              // 16-tall row tile
  if (tN * 16 >= Ncol || tM * 16 >= M) return;   // wave-uniform
  v8f acc = wmma_dotT_f32(A, Bw, K, tM * 16, tN * 16);
  const int lane = threadIdx.x & 31;
  const int half = lane >> 4;
  const int l    = lane & 15;
  const int gc = tN * 16 + l;
  const float bb = bias[gc];
  #pragma unroll
  for (int r = 0; r < 8; ++r) {
    const int gr = tM * 16 + 8 * half + r;
    float v = acc[r] + bb;
    if (relu) v = fmaxf(v, 0.0f);
    C[(size_t)gr * Ncol + gc] = v;
  }
}

// dvec[row] = sum_k e[row][k]^2   (row = b*N + i, flat over 4096 rows)
__global__ void k_rowsq(const float* __restrict__ e, float* __restrict__ dvec,
                        int rows, int E) {
  int i = blockIdx.x * blockDim.x + threadIdx.x;
  if (i >= rows) return;
  const float* p = e + (size_t)i * E;
  float s = 0.f;
  for (int k = 0; k < E; ++k) s += p[k] * p[k];
  dvec[i] = s;
}

// P[b][i][j] = (1 - corr/denom)/2, corr = e_b @ e_b^T via WMMA
__global__ void k_corrD(const float* __restrict__ e, const float* __restrict__ dvec,
                        float* __restrict__ P, int N, int E) {
  const int b = blockIdx.z;
  const float* eb = e + (size_t)b * N * E;
  const float* db = dvec + b * N;
  float* Pb = P + (size_t)b * N * N;
  const int wave = threadIdx.x >> 5;
  const int tJ = blockIdx.x * 8 + wave;
  const int tI = blockIdx.y;
  v8f acc = wmma_dotT_f32(eb, eb, E, tI * 16, tJ * 16);
  const int lane = threadIdx.x & 31;
  const int half = lane >> 4;
  const int l    = lane & 15;
  const int gc = tJ * 16 + l;
  const float dj = db[gc];
  #pragma unroll
  for (int r = 0; r < 8; ++r) {
    const int gr = tI * 16 + 8 * half + r;
    const float di = db[gr];
    const float denom = sqrtf(fmaxf(di * dj, 1e-8f));
    Pb[(size_t)gr * N + gc] = 0.5f * (1.0f - acc[r] / denom);
  }
}

// In-place inclusive row scan (one block per row). N must be 2048 (256 thr x 8)
__global__ void k_rowscan(float* __restrict__ P, int N) {
  const int b = blockIdx.y;
  const int row = blockIdx.x;
  float* p = P + ((size_t)b * N + row) * N;
  __shared__ float part[256];
  const int t = threadIdx.x;
  const int per = N / 256;   // 8
  float vals[8];
  float s = 0.f;
  for (int k = 0; k < per; ++k) { s += p[t * per + k]; vals[k] = s; }
  part[t] = s;
  __syncthreads();
  for (int off = 1; off < 256; off <<= 1) {
    float x = (t >= off) ? part[t - off] : 0.f;
    __syncthreads();
    part[t] += x;
    __syncthreads();
  }
  const float base = (t > 0) ? part[t - 1] : 0.f;
  for (int k = 0; k < per; ++k) p[t * per + k] = base + vals[k];
}

// In-place inclusive column scan: one thread per column, coalesced across j
__global__ void k_colscan(float* __restrict__ P, int N) {
  const int b = blockIdx.y;
  const int j = blockIdx.x * blockDim.x + threadIdx.x;
  if (j >= N) return;
  float* Pb = P + (size_t)b * N * N;
  float run = 0.f;
  for (int i = 0; i < N; ++i) {
    run += Pb[(size_t)i * N + j];
    Pb[(size_t)i * N + j] = run;
  }
}

// dP[b][j] = P[b][j][j]
__global__ void k_diag(const float* __restrict__ P, float* __restrict__ dP, int N) {
  const int b = blockIdx.y;
  const int j = blockIdx.x * blockDim.x + threadIdx.x;
  if (j >= N) return;
  dP[b * N + j] = P[(size_t)b * N * N + (size_t)j * N + j];
}

// C0[i] = Bsum[i][N-1]; tsum init (N at j=N-1, else 0).
// Bsum[i][j] = dP[j] - 2*P[i-1][j] + dP[i-1]   (i>0; P symmetric),  dP[j] for i==0
__global__ void k_init(const float* __restrict__ P, const float* __restrict__ dP,
                       float* __restrict__ C0, float* __restrict__ tsum, int N) {
  const int b = blockIdx.y;
  const int i = blockIdx.x * blockDim.x + threadIdx.x;
  if (i >= N) return;
  const float* Pb  = P  + (size_t)b * N * N;
  const float* dPb = dP + b * N;
  float v = dPb[N - 1];
  if (i > 0) v += dPb[i - 1] - 2.f * Pb[(size_t)(i - 1) * N + (N - 1)];
  C0[b * N + i] = v;
  tsum[b * N + i] = (i == N - 1) ? (float)N : 0.f;
}

// DP row pass (iteration kk): for row i < N-kk compute
//   Cnew[i] = min_{j in [i, N-kk)} M[i][j],  invZ[i] = 1 / sum_j exp(Cnew[i]-M[i][j])
// M[i][j] = Bsum[i][j] + Cprev[j+1].  M row staged in LDS (read Bsum once).
__global__ void k_row(const float* __restrict__ P, const float* __restrict__ dP,
                      const float* __restrict__ Cprev, float* __restrict__ Cnew,
                      float* __restrict__ invZ, int N, int kk) {
  const int b = blockIdx.y;
  const int i = blockIdx.x;
  const int limit = N - kk;
  const int t = threadIdx.x;
  if (i >= limit) {
    if (t == 0) { Cnew[b * N + i] = 0.f; invZ[b * N + i] = 0.f; }
    return;
  }
  const float* Pb  = P  + (size_t)b * N * N;
  const float* dPb = dP + b * N;
  const float* Cp  = Cprev + b * N;
  __shared__ float mrow[NSEQ];
  __shared__ float red[256];
  const float dPi = (i > 0) ? dPb[i - 1] : 0.f;
  const float* Prow = Pb + (size_t)(i > 0 ? i - 1 : 0) * N;
  float lmin = BIGF;
  for (int j = i + t; j < limit; j += 256) {
    float bs = dPb[j] + ((i > 0) ? (dPi - 2.f * Prow[j]) : 0.f);
    float m = bs + Cp[j + 1];
    mrow[j] = m;
    lmin = fminf(lmin, m);
  }
  red[t] = lmin;
  __syncthreads();
  for (int off = 128; off > 0; off >>= 1) {
    if (t < off) red[t] = fminf(red[t], red[t + off]);
    __syncthreads();
  }
  const float rmin = red[0];
  __syncthreads();
  float ls = 0.f;
  for (int j = i + t; j < limit; j += 256) ls += __expf(rmin - mrow[j]);
  red[t] = ls;
  __syncthreads();
  for (int off = 128; off > 0; off >>= 1) {
    if (t < off) red[t] += red[t + off];
    __syncthreads();
  }
  if (t == 0) { Cnew[b * N + i] = rmin; invZ[b * N + i] = 1.f / red[0]; }
}

// DP column pass: tsum[j] += sum_{i=0..j} exp(Cnew[i]-M[i][j]) * invZ[i]
// (j < N-kk). One thread per column; P loads coalesced across j; per-i scalars
// (Cnew[i], dP[i-1], invZ[i]) are wave-uniform broadcasts. Deterministic.
__global__ void k_col(const float* __restrict__ P, const float* __restrict__ dP,
                      const float* __restrict__ Cprev, const float* __restrict__ Cnew,
                      const float* __restrict__ invZ, float* __restrict__ tsum,
                      int N, int kk) {
  const int b = blockIdx.y;
  const int j = blockIdx.x * blockDim.x + threadIdx.x;
  const int limit = N - kk;
  if (j >= limit) return;
  const float* Pb  = P  + (size_t)b * N * N;
  const float* dPb = dP + b * N;
  const float* Cp  = Cprev + b * N;
  const float* Cn  = Cnew  + b * N;
  const float* iz  = invZ  + b * N;
  const float dPj = dPb[j];
  const float csh = Cp[j + 1];
  float acc = __expf(Cn[0] - (dPj + csh)) * iz[0];         // i == 0
  for (int i = 1; i <= j; ++i) {
    float m = dPj - 2.f * Pb[(size_t)(i - 1) * N + j] + dPb[i - 1] + csh;
    acc += __expf(Cn[i] - m) * iz[i];
  }
  tsum[b * N + j] += acc;
}

// out[b][j] = tsum[b][j] / tcnt[j]; tcnt closed-form (data-independent)
__global__ void k_final(const float* __restrict__ tsum, float* __restrict__ out,
                        int N, int Ksegs) {
  const int b = blockIdx.y;
  const int j = blockIdx.x * blockDim.x + threadIdx.x;
  if (j >= N) return;
  const int cap = N - 1 - j;
  const int reps = (Ksegs - 1) < cap ? (Ksegs - 1) : cap;
  float cnt = (float)(j + 1) * (float)reps;
  if (j == N - 1) cnt += (float)N;
  out[b * N + j] = tsum[b * N + j] / cnt;
}

// ---------------------------------------------------------------------------
extern "C" void kernel_launch(void* const* d_in, const int* in_sizes, int n_in,
                              void* d_out, int out_size, void* d_ws, size_t ws_size,
                              hipStream_t stream) {
  const float* x  = (const float*)d_in[0];   // (2,2048,1024)
  const float* W0 = (const float*)d_in[1];   // (512,1024)
  const float* b0 = (const float*)d_in[2];   // (512,)
  const float* W1 = (const float*)d_in[3];   // (256,512)
  const float* b1 = (const float*)d_in[4];   // (256,)
  float* out = (float*)d_out;                // (2,2048)

  float* ws = (float*)d_ws;
  size_t o = 0;
  float* P    = ws + o; o += (size_t)BB * NSEQ * NSEQ;  // 33.6 MB: D -> prefix P
  float* h    = ws + o; o += (size_t)MROWS * HH;        // 8 MB
  float* e    = ws + o; o += (size_t)MROWS * EE;        // 4 MB
  float* dvec = ws + o; o += (size_t)BB * NSEQ;
  float* dP   = ws + o; o += (size_t)BB * NSEQ;
  float* Ca   = ws + o; o += (size_t)BB * NSEQ;
  float* Cb   = ws + o; o += (size_t)BB * NSEQ;
  float* iZ   = ws + o; o += (size_t)BB * NSEQ;
  float* ts   = ws + o; o += (size_t)BB * NSEQ;

  const dim3 blk(256);

  // h = relu(x @ W0^T + b0): M=4096, Ncol=512, K=1024 -> 32 col tiles / 8 waves
  k_gemm<<<dim3(HH / 16 / 8, MROWS / 16, 1), blk, 0, stream>>>(
      x, W0, b0, h, MROWS, HH, FF, 1);
  // e = h @ W1^T + b1: M=4096, Ncol=256, K=512
  k_gemm<<<dim3(EE / 16 / 8, MROWS / 16, 1), blk, 0, stream>>>(
      h, W1, b1, e, MROWS, EE, HH, 0);
  // diag energies
  k_rowsq<<<dim3(MROWS / 256), blk, 0, stream>>>(e, dvec, MROWS, EE);
  // cosine-distance matrix D (written into P buffer), corr via WMMA
  k_corrD<<<dim3(NSEQ / 16 / 8, NSEQ / 16, BB), blk, 0, stream>>>(e, dvec, P, NSEQ, EE);
  // 2D inclusive prefix sum of D: rows then columns (in place)
  k_rowscan<<<dim3(NSEQ, BB), blk, 0, stream>>>(P, NSEQ);
  k_colscan<<<dim3(NSEQ / 256, BB), blk, 0, stream>>>(P, NSEQ);
  k_diag<<<dim3(NSEQ / 256, BB), blk, 0, stream>>>(P, dP, NSEQ);
  // C0 = Bsum[:, N-1]; tsum init
  k_init<<<dim3(NSEQ / 256, BB), blk, 0, stream>>>(P, dP, Ca, ts, NSEQ);

  // 29 sequentially-dependent DP steps
  float* Cp = Ca;
  float* Cn = Cb;
  for (int kk = 1; kk < KSEG; ++kk) {
    k_row<<<dim3(NSEQ, BB), blk, 0, stream>>>(P, dP, Cp, Cn, iZ, NSEQ, kk);
    k_col<<<dim3(NSEQ / 256, BB), blk, 0, stream>>>(P, dP, Cp, Cn, iZ, ts, NSEQ, kk);
    float* t2 = Cp; Cp = Cn; Cn = t2;
  }

  k_final<<<dim3(NSEQ / 256, BB), blk, 0, stream>>>(ts, out, NSEQ, KSEG);

  (void)in_sizes; (void)n_in; (void)out_size; (void)ws_size;
}